// GatedGCNNet_16793322127657
// MI455X (gfx1250) — compile-verified
//
#include <hip/hip_runtime.h>

#define NNODES 100000
#define NEDGES 1600000
#define BB 4
#define INC 128
#define HIDC 16
#define OUTC 64
#define ROWS (BB * NNODES)      // 400000 rows (b*N+n)
#define TILES (ROWS / 16)       // 25000 16-row WMMA tiles

typedef float v2f __attribute__((ext_vector_type(2)));
typedef float v8f __attribute__((ext_vector_type(8)));

// D(16x16,f32) = A(16x4,f32) * B(4x16,f32) + C  -- full-precision CDNA5 WMMA
static __device__ __forceinline__ v8f wmma4(v2f a, v2f b, v8f c) {
  return __builtin_amdgcn_wmma_f32_16x16x4_f32(false, a, false, b, (short)0, c,
                                               false, false);
}

__global__ void k_zero(float* __restrict__ p, int n) {
  int i = blockIdx.x * blockDim.x + threadIdx.x;
  if (i < n) p[i] = 0.0f;
}

// x1[r, 0:16] = X[b, n_id[n], 0:128] @ W1 (128x16); r = b*N+n
__global__ void k_gemm_in(const float* __restrict__ X,
                          const int* __restrict__ n_id,
                          const float* __restrict__ W1,
                          float* __restrict__ out) {
  int wave = (blockIdx.x * blockDim.x + threadIdx.x) >> 5;
  int lane = threadIdx.x & 31;
  if (wave >= TILES) return;
  int m = lane & 15, kh = lane >> 4;
  int r = wave * 16 + m;
  int b = r / NNODES, n = r - b * NNODES;
  const float* arow = X + ((size_t)b * NNODES + n_id[n]) * INC;
  v8f acc = {};
  for (int k0 = 0; k0 < INC; k0 += 4) {
    v2f a, bm;
    a.x = arow[k0 + 2 * kh];
    a.y = arow[k0 + 2 * kh + 1];
    bm.x = W1[(k0 + 2 * kh) * HIDC + m];
    bm.y = W1[(k0 + 2 * kh + 1) * HIDC + m];
    acc = wmma4(a, bm, acc);
  }
  for (int i = 0; i < 8; ++i)
    out[(size_t)(wave * 16 + i + 8 * kh) * HIDC + m] = acc[i];
}

// xU[r] = x1[b, res[n]] @ U ;  xV[r] = x1[r] @ V   (16x16 each)
__global__ void k_dual16(const float* __restrict__ x1,
                         const int* __restrict__ res,
                         const float* __restrict__ U,
                         const float* __restrict__ V,
                         float* __restrict__ xU, float* __restrict__ xV) {
  int wave = (blockIdx.x * blockDim.x + threadIdx.x) >> 5;
  int lane = threadIdx.x & 31;
  if (wave >= TILES) return;
  int m = lane & 15, kh = lane >> 4;
  int r = wave * 16 + m;
  int b = r / NNODES, n = r - b * NNODES;
  const float* aV = x1 + (size_t)r * HIDC;
  const float* aU = x1 + ((size_t)b * NNODES + res[n]) * HIDC;
  v8f accU = {}, accV = {};
  for (int k0 = 0; k0 < HIDC; k0 += 4) {
    v2f au, av, bu, bv;
    au.x = aU[k0 + 2 * kh];        au.y = aU[k0 + 2 * kh + 1];
    av.x = aV[k0 + 2 * kh];        av.y = aV[k0 + 2 * kh + 1];
    bu.x = U[(k0 + 2 * kh) * HIDC + m]; bu.y = U[(k0 + 2 * kh + 1) * HIDC + m];
    bv.x = V[(k0 + 2 * kh) * HIDC + m]; bv.y = V[(k0 + 2 * kh + 1) * HIDC + m];
    accU = wmma4(au, bu, accU);
    accV = wmma4(av, bv, accV);
  }
  for (int i = 0; i < 8; ++i) {
    size_t ro = (size_t)(wave * 16 + i + 8 * kh) * HIDC + m;
    xU[ro] = accU[i];
    xV[ro] = accV[i];
  }
}

// x2[r, 0:64] = o1[r, 0:16] @ W (16x64)
__global__ void k_gemm_h2o(const float* __restrict__ x,
                           const float* __restrict__ W,
                           float* __restrict__ out) {
  int wave = (blockIdx.x * blockDim.x + threadIdx.x) >> 5;
  int lane = threadIdx.x & 31;
  if (wave >= TILES) return;
  int m = lane & 15, kh = lane >> 4;
  const float* arow = x + (size_t)(wave * 16 + m) * HIDC;
  v8f acc[4] = {{}, {}, {}, {}};
  for (int k0 = 0; k0 < HIDC; k0 += 4) {
    v2f a;
    a.x = arow[k0 + 2 * kh];
    a.y = arow[k0 + 2 * kh + 1];
    for (int nt = 0; nt < 4; ++nt) {
      v2f bm;
      bm.x = W[(k0 + 2 * kh) * OUTC + nt * 16 + m];
      bm.y = W[(k0 + 2 * kh + 1) * OUTC + nt * 16 + m];
      acc[nt] = wmma4(a, bm, acc[nt]);
    }
  }
  for (int nt = 0; nt < 4; ++nt)
    for (int i = 0; i < 8; ++i)
      out[(size_t)(wave * 16 + i + 8 * kh) * OUTC + nt * 16 + m] = acc[nt][i];
}

// xU[r] = x2[b,res[n]] @ U (64x64); xV[r] = x2[r] @ V (64x64)
__global__ void k_dual64(const float* __restrict__ x2,
                         const int* __restrict__ res,
                         const float* __restrict__ U,
                         const float* __restrict__ V,
                         float* __restrict__ xU, float* __restrict__ xV) {
  int wave = (blockIdx.x * blockDim.x + threadIdx.x) >> 5;
  int lane = threadIdx.x & 31;
  if (wave >= TILES) return;
  int m = lane & 15, kh = lane >> 4;
  int r = wave * 16 + m;
  int b = r / NNODES, n = r - b * NNODES;
  const float* aV = x2 + (size_t)r * OUTC;
  const float* aU = x2 + ((size_t)b * NNODES + res[n]) * OUTC;
  v8f accU[4] = {{}, {}, {}, {}};
  v8f accV[4] = {{}, {}, {}, {}};
  for (int k0 = 0; k0 < OUTC; k0 += 4) {
    v2f au, av;
    au.x = aU[k0 + 2 * kh]; au.y = aU[k0 + 2 * kh + 1];
    av.x = aV[k0 + 2 * kh]; av.y = aV[k0 + 2 * kh + 1];
    for (int nt = 0; nt < 4; ++nt) {
      v2f bu, bv;
      bu.x = U[(k0 + 2 * kh) * OUTC + nt * 16 + m];
      bu.y = U[(k0 + 2 * kh + 1) * OUTC + nt * 16 + m];
      bv.x = V[(k0 + 2 * kh) * OUTC + nt * 16 + m];
      bv.y = V[(k0 + 2 * kh + 1) * OUTC + nt * 16 + m];
      accU[nt] = wmma4(au, bu, accU[nt]);
      accV[nt] = wmma4(av, bv, accV[nt]);
    }
  }
  for (int nt = 0; nt < 4; ++nt)
    for (int i = 0; i < 8; ++i) {
      size_t ro = (size_t)(wave * 16 + i + 8 * kh) * OUTC + nt * 16 + m;
      xU[ro] = accU[nt][i];
      xV[ro] = accV[nt][i];
    }
}

__global__ void k_cnt(const int* __restrict__ ei, float* __restrict__ cnt) {
  int e = blockIdx.x * blockDim.x + threadIdx.x;
  if (e < NEDGES) atomicAdd(&cnt[ei[NEDGES + e]], 1.0f);
}

// sums[b,dst,c] += (ew[e]*W2[c]) * xV[b,src,c]   (C channels per edge)
template <int C>
__global__ void k_scatter(const int* __restrict__ ei,
                          const float* __restrict__ ew,
                          const float* __restrict__ W2,
                          const float* __restrict__ xV,
                          float* __restrict__ sums) {
  int gid = blockIdx.x * blockDim.x + threadIdx.x;
  int e = gid / C, c = gid % C;
  if (e >= NEDGES) return;
  int src = ei[e], dst = ei[NEDGES + e];
  float emb = ew[e] * W2[c];
  for (int b = 0; b < BB; ++b) {
    float v = emb * xV[((size_t)b * NNODES + src) * C + c];
    atomicAdd(&sums[((size_t)b * NNODES + dst) * C + c], v);
  }
}

// a = xU + sums/max(cnt,1); LayerNorm over (B,C) per node; out = xr + relu(a)
// One wave32 per node; VPL values per lane (VPL = B*C/32).
template <int C, int VPL>
__global__ void k_finalize(const float* __restrict__ xU,
                           const float* __restrict__ sums,
                           const float* __restrict__ cnt,
                           const float* __restrict__ x,
                           const int* __restrict__ res,
                           float* __restrict__ out) {
  int n = (blockIdx.x * blockDim.x + threadIdx.x) >> 5;
  int lane = threadIdx.x & 31;
  if (n >= NNODES) return;
  float rc = 1.0f / fmaxf(cnt[n], 1.0f);
  int rn = res[n];
  float a[VPL], xr[VPL];
  float s = 0.0f, sq = 0.0f;
  for (int j = 0; j < VPL; ++j) {
    int v = lane + 32 * j;
    int b = v / C, c = v % C;
    size_t idx = ((size_t)b * NNODES + n) * C + c;
    float val = xU[idx] + sums[idx] * rc;
    a[j] = val;
    xr[j] = x[((size_t)b * NNODES + rn) * C + c];
    s += val;
    sq += val * val;
  }
  for (int msk = 16; msk >= 1; msk >>= 1) {
    s += __shfl_xor(s, msk, 32);
    sq += __shfl_xor(sq, msk, 32);
  }
  const float inv = 1.0f / (float)(BB * C);
  float mean = s * inv;
  float var = sq * inv - mean * mean;
  float rstd = rsqrtf(var + 1e-5f);
  for (int j = 0; j < VPL; ++j) {
    int v = lane + 32 * j;
    int b = v / C, c = v % C;
    float an = fmaxf((a[j] - mean) * rstd, 0.0f);
    out[((size_t)b * NNODES + n) * C + c] = xr[j] + an;
  }
}

extern "C" void kernel_launch(void* const* d_in, const int* in_sizes, int n_in,
                              void* d_out, int out_size, void* d_ws,
                              size_t ws_size, hipStream_t stream) {
  (void)in_sizes; (void)n_in; (void)out_size; (void)ws_size;
  const float* X    = (const float*)d_in[0];
  const int*   ei0  = (const int*)d_in[1];
  const float* ew0  = (const float*)d_in[2];
  const int*   ei1  = (const int*)d_in[3];
  const float* ew1  = (const float*)d_in[4];
  const int*   n_id = (const int*)d_in[5];
  const int*   res0 = (const int*)d_in[6];
  const int*   res1 = (const int*)d_in[7];
  const float* W1_1 = (const float*)d_in[8];
  const float* W2_1 = (const float*)d_in[9];
  const float* U1   = (const float*)d_in[10];
  const float* V1   = (const float*)d_in[11];
  const float* W1_2 = (const float*)d_in[12];
  const float* W2_2 = (const float*)d_in[13];
  const float* U2   = (const float*)d_in[14];
  const float* V2   = (const float*)d_in[15];

  const size_t NH = (size_t)ROWS * HIDC;  // 6.4M floats
  const size_t NO = (size_t)ROWS * OUTC;  // 25.6M floats
  float* ws   = (float*)d_ws;
  float* x1   = ws;            // layer-1 hidden (also xr for layer 1)
  float* xU1  = x1 + NH;
  float* xV1  = xU1 + NH;
  float* s1   = xV1 + NH;
  float* o1   = s1 + NH;       // layer-1 output / layer-2 input
  float* cnt1 = o1 + NH;
  float* cnt2 = cnt1 + NNODES;
  float* x2   = cnt2 + NNODES; // layer-2 hidden (xr for layer 2)
  float* xU2  = x2 + NO;
  float* xV2  = xU2 + NO;
  float* s2   = xV2 + NO;      // total ~134.6M floats (~539 MB)

  const int T = 256;
  // zero accumulators (harness poisons ws; must re-zero every call)
  k_zero<<<(int)((NH + T - 1) / T), T, 0, stream>>>(s1, (int)NH);
  k_zero<<<(int)((NO + T - 1) / T), T, 0, stream>>>(s2, (int)NO);
  k_zero<<<(NNODES + T - 1) / T, T, 0, stream>>>(cnt1, NNODES);
  k_zero<<<(NNODES + T - 1) / T, T, 0, stream>>>(cnt2, NNODES);

  const int gemmBlocks = TILES * 32 / T;  // 3125

  // ---- layer 1 (128 -> 16) ----
  k_gemm_in<<<gemmBlocks, T, 0, stream>>>(X, n_id, W1_1, x1);
  k_dual16<<<gemmBlocks, T, 0, stream>>>(x1, res0, U1, V1, xU1, xV1);
  k_cnt<<<NEDGES / T, T, 0, stream>>>(ei0, cnt1);
  k_scatter<HIDC><<<NEDGES * HIDC / T, T, 0, stream>>>(ei0, ew0, W2_1, xV1, s1);
  k_finalize<HIDC, 2><<<NNODES * 32 / T, T, 0, stream>>>(xU1, s1, cnt1, x1,
                                                         res0, o1);

  // ---- layer 2 (16 -> 64) ----
  k_gemm_h2o<<<gemmBlocks, T, 0, stream>>>(o1, W1_2, x2);
  k_dual64<<<gemmBlocks, T, 0, stream>>>(x2, res1, U2, V2, xU2, xV2);
  k_cnt<<<NEDGES / T, T, 0, stream>>>(ei1, cnt2);
  k_scatter<OUTC><<<NEDGES * OUTC / T, T, 0, stream>>>(ei1, ew1, W2_2, xV2, s2);
  k_finalize<OUTC, 8><<<NNODES * 32 / T, T, 0, stream>>>(xU2, s2, cnt2, x2,
                                                         res1, (float*)d_out);
}